// NTMMemory_14439680049629
// MI455X (gfx1250) — compile-verified
//
#include <hip/hip_runtime.h>
#include <math.h>

// NTM memory addressing + read/write, fused, one workgroup per batch.
// B=128, N=8192, M=64. f32 throughout (matches reference dtype).
// WMMA use: V_WMMA_F32_16X16X4_F32 computes the content-addressing GEMV
// dot[n] = sum_m (mem[n,m]+eps)*(k[m]+eps) exactly in f32.
// Cache policy: phase-1 memory reads RT (hope for L2 reuse in phase 4);
// phase-4 memory reads NT (last use); all big output stores NT (streaming).

typedef __attribute__((ext_vector_type(2))) float v2f;
typedef __attribute__((ext_vector_type(4))) float v4f;
typedef __attribute__((ext_vector_type(8))) float v8f;

#define EPSF      1e-16f
#define COS_EPSF  1e-8f
#define BATCH     128
#define NN        8192
#define MM        64
#define NTHREADS  1024
#define NWAVES    (NTHREADS / 32)          // 32
#define ROWS_PER_WAVE (NN / NWAVES)        // 256
#define CHUNKS_PER_WAVE (ROWS_PER_WAVE/16) // 16

__global__ __launch_bounds__(NTHREADS)
void ntm_fused_kernel(const float* __restrict__ memory,
                      const float* __restrict__ k,
                      const float* __restrict__ beta_p,
                      const float* __restrict__ g_p,
                      const float* __restrict__ s_p,
                      const float* __restrict__ gamma_p,
                      const float* __restrict__ w_prev,
                      const float* __restrict__ e_p,
                      const float* __restrict__ a_p,
                      float* __restrict__ out_w,
                      float* __restrict__ out_r,
                      float* __restrict__ out_m)
{
    __shared__ __align__(16) float lds_w[NN];   // 32 KB working array for w
    __shared__ float lds_red[NWAVES + 1];
    __shared__ float lds_k[MM];

    const int b    = blockIdx.x;
    const int tid  = threadIdx.x;
    const int wave = tid >> 5;
    const int lane = tid & 31;

    const float beta  = beta_p[b];
    const float g     = g_p[b];
    const float s0    = s_p[3*b + 0];
    const float s1    = s_p[3*b + 1];
    const float s2    = s_p[3*b + 2];
    const float gamma = gamma_p[b];

    // ---- phase 0: stage k+eps in LDS, compute ||k+eps|| --------------------
    if (tid < MM) lds_k[tid] = k[(size_t)b*MM + tid] + EPSF;
    __syncthreads();
    if (tid == 0) {
        float s = 0.f;
        #pragma unroll
        for (int m = 0; m < MM; ++m) s = fmaf(lds_k[m], lds_k[m], s);
        lds_red[NWAVES] = sqrtf(s);
    }
    __syncthreads();
    const float norm_k = lds_red[NWAVES];

    // ---- phase 1: WMMA f32 content scoring ---------------------------------
    // A (16x4): lane 0-15 -> row=lane, K={0,1}; lane 16-31 -> row=lane-16, K={2,3}
    // B (4x16): mirror layout; every column loaded with the same k values, so
    // D[i][j] = dot(row i, k) for every j (exact f32).
    {
        const int rsel = lane & 15;
        const int koff = (lane >> 4) * 2;
        const size_t mem_base = ((size_t)b * NN) * MM;
        for (int ch = 0; ch < CHUNKS_PER_WAVE; ++ch) {
            const int base_n = wave * ROWS_PER_WAVE + ch * 16;
            const float* rowptr = memory + mem_base + (size_t)(base_n + rsel) * MM;
            v8f acc = {};
            float sq = 0.f;
            #pragma unroll
            for (int it = 0; it < 16; ++it) {       // K = 64 in steps of 4
                const int m0 = it*4 + koff;
                v2f av, bv;
                av.x = rowptr[m0]     + EPSF;
                av.y = rowptr[m0 + 1] + EPSF;
                bv.x = lds_k[m0];
                bv.y = lds_k[m0 + 1];
                sq = fmaf(av.x, av.x, fmaf(av.y, av.y, sq));
                acc = __builtin_amdgcn_wmma_f32_16x16x4_f32(
                        false, av, false, bv, (short)0, acc, false, false);
            }
            // row sumsq: lane l and l+16 each hold half of row (l&15)
            sq += __shfl_xor(sq, 16, 32);
            // lanes 0..7 emit rows 0..7 (own norm); lanes 16..23 emit rows
            // 8..15 (norm held by lane row = lane-8)
            const int  src    = (lane < 16) ? lane : (lane - 8);
            const float sqrow = __shfl(sq, src, 32);
            const int  q      = lane & 7;
            float dotv = acc[0];
            #pragma unroll
            for (int p = 1; p < 8; ++p) if (q == p) dotv = acc[p];
            const bool writer = (lane < 8) || (lane >= 16 && lane < 24);
            if (writer) {
                const int row_out = (lane < 8) ? lane : (lane - 8);
                const float denom = fmaxf(sqrtf(sqrow) * norm_k, COS_EPSF);
                lds_w[base_n + row_out] = beta * (dotv / denom);
            }
        }
    }
    __syncthreads();

    // ---- phase 2: softmax over N + gate interpolation ----------------------
    float val[8];
    #pragma unroll
    for (int i = 0; i < 8; ++i) val[i] = lds_w[i*NTHREADS + tid];

    float mx = val[0];
    #pragma unroll
    for (int i = 1; i < 8; ++i) mx = fmaxf(mx, val[i]);
    #pragma unroll
    for (int off = 16; off > 0; off >>= 1) mx = fmaxf(mx, __shfl_xor(mx, off, 32));
    if (lane == 0) lds_red[wave] = mx;
    __syncthreads();
    if (wave == 0) {
        float t = lds_red[lane];
        #pragma unroll
        for (int off = 16; off > 0; off >>= 1) t = fmaxf(t, __shfl_xor(t, off, 32));
        if (lane == 0) lds_red[NWAVES] = t;
    }
    __syncthreads();
    const float maxv = lds_red[NWAVES];

    float sum = 0.f;
    #pragma unroll
    for (int i = 0; i < 8; ++i) { val[i] = expf(val[i] - maxv); sum += val[i]; }
    #pragma unroll
    for (int off = 16; off > 0; off >>= 1) sum += __shfl_xor(sum, off, 32);
    if (lane == 0) lds_red[wave] = sum;
    __syncthreads();
    if (wave == 0) {
        float t = lds_red[lane];
        #pragma unroll
        for (int off = 16; off > 0; off >>= 1) t += __shfl_xor(t, off, 32);
        if (lane == 0) lds_red[NWAVES] = t;
    }
    __syncthreads();
    const float inv_sum = 1.f / lds_red[NWAVES];

    #pragma unroll
    for (int i = 0; i < 8; ++i) {
        const int n = i*NTHREADS + tid;
        const float wr = val[i] * inv_sum;
        lds_w[n] = g * wr + (1.f - g) * w_prev[(size_t)b*NN + n];
    }
    __syncthreads();

    // ---- phase 3: circular 3-tap shift + sharpen + renormalize -------------
    float wp[8];
    float sum2 = 0.f;
    #pragma unroll
    for (int i = 0; i < 8; ++i) {
        const int n = i*NTHREADS + tid;
        const float sh = lds_w[(n + NN - 1) & (NN - 1)] * s0
                       + lds_w[n]                        * s1
                       + lds_w[(n + 1) & (NN - 1)]       * s2;
        wp[i] = powf(sh, gamma);
        sum2 += wp[i];
    }
    #pragma unroll
    for (int off = 16; off > 0; off >>= 1) sum2 += __shfl_xor(sum2, off, 32);
    if (lane == 0) lds_red[wave] = sum2;
    __syncthreads();
    if (wave == 0) {
        float t = lds_red[lane];
        #pragma unroll
        for (int off = 16; off > 0; off >>= 1) t += __shfl_xor(t, off, 32);
        if (lane == 0) lds_red[NWAVES] = t;
    }
    __syncthreads();
    const float inv_w = 1.f / (lds_red[NWAVES] + EPSF);

    #pragma unroll
    for (int i = 0; i < 8; ++i) {
        const int n = i*NTHREADS + tid;
        const float wf = wp[i] * inv_w;
        lds_w[n] = wf;
        __builtin_nontemporal_store(wf, out_w + (size_t)b*NN + n);  // streaming
    }
    __syncthreads();

    // ---- phase 4: fused read (r = w^T mem) + erase/add write ---------------
    const int mgrp = tid & 15;       // 16 float4 slices across M=64
    const int ngrp = tid >> 4;       // 64 row groups
    const int m0   = mgrp * 4;
    const float e0 = e_p[(size_t)b*MM + m0 + 0];
    const float e1 = e_p[(size_t)b*MM + m0 + 1];
    const float e2 = e_p[(size_t)b*MM + m0 + 2];
    const float e3 = e_p[(size_t)b*MM + m0 + 3];
    const float a0 = a_p[(size_t)b*MM + m0 + 0];
    const float a1 = a_p[(size_t)b*MM + m0 + 1];
    const float a2 = a_p[(size_t)b*MM + m0 + 2];
    const float a3 = a_p[(size_t)b*MM + m0 + 3];
    const float* memb = memory + (size_t)b*NN*MM;
    float*       outb = out_m  + (size_t)b*NN*MM;

    float r0 = 0.f, r1 = 0.f, r2 = 0.f, r3 = 0.f;
    for (int i = 0; i < NN/64; ++i) {
        const int n = i*64 + ngrp;
        const float wn = lds_w[n];
        // last-use read of this memory line: non-temporal
        const v4f mv = __builtin_nontemporal_load(
                           (const v4f*)(memb + (size_t)n*MM + m0));
        r0 = fmaf(wn, mv.x, r0);
        r1 = fmaf(wn, mv.y, r1);
        r2 = fmaf(wn, mv.z, r2);
        r3 = fmaf(wn, mv.w, r3);
        v4f ov;
        ov.x = mv.x * (1.f - wn*e0) + wn*a0;
        ov.y = mv.y * (1.f - wn*e1) + wn*a1;
        ov.z = mv.z * (1.f - wn*e2) + wn*a2;
        ov.w = mv.w * (1.f - wn*e3) + wn*a3;
        // pure streaming store: non-temporal, don't pollute L2
        __builtin_nontemporal_store(ov, (v4f*)(outb + (size_t)n*MM + m0));
    }
    __syncthreads();   // all lds_w reads done; reuse lds_w as v4f scratch

    v4f* lds4 = (v4f*)lds_w;
    v4f rr; rr.x = r0; rr.y = r1; rr.z = r2; rr.w = r3;
    lds4[tid] = rr;
    __syncthreads();
    // deterministic tree reduce over the 64 row-groups (same mgrp lanes)
    for (int s = 512; s >= 16; s >>= 1) {
        if (tid < s) {
            v4f x = lds4[tid];
            v4f y = lds4[tid + s];
            x = x + y;
            lds4[tid] = x;
        }
        __syncthreads();
    }
    if (tid < 16) {
        const v4f x = lds4[tid];
        float* orow = out_r + (size_t)b*MM + tid*4;
        __builtin_nontemporal_store(x.x, orow + 0);
        __builtin_nontemporal_store(x.y, orow + 1);
        __builtin_nontemporal_store(x.z, orow + 2);
        __builtin_nontemporal_store(x.w, orow + 3);
    }
}

extern "C" void kernel_launch(void* const* d_in, const int* in_sizes, int n_in,
                              void* d_out, int out_size, void* d_ws, size_t ws_size,
                              hipStream_t stream) {
    (void)in_sizes; (void)n_in; (void)out_size; (void)d_ws; (void)ws_size;
    const float* memory = (const float*)d_in[0];
    const float* k      = (const float*)d_in[1];
    const float* beta   = (const float*)d_in[2];
    const float* g      = (const float*)d_in[3];
    const float* s      = (const float*)d_in[4];
    const float* gamma  = (const float*)d_in[5];
    const float* w_prev = (const float*)d_in[6];
    const float* e      = (const float*)d_in[7];
    const float* a      = (const float*)d_in[8];

    float* out_w = (float*)d_out;                          // [B, N]
    float* out_r = out_w + (size_t)BATCH * NN;             // [B, M]
    float* out_m = out_r + (size_t)BATCH * MM;             // [B, N, M]

    ntm_fused_kernel<<<dim3(BATCH), dim3(NTHREADS), 0, stream>>>(
        memory, k, beta, g, s, gamma, w_prev, e, a, out_w, out_r, out_m);
}